// StickyRNNModel_28363964023163
// MI455X (gfx1250) — compile-verified
//
#include <hip/hip_runtime.h>

typedef __attribute__((ext_vector_type(2))) float v2f;
typedef __attribute__((ext_vector_type(8))) float v8f;

constexpr int Bdim = 2048;
constexpr int Tdim = 1024;
constexpr int VOC  = 65;   // 26+26+10+3
constexpr int EMB  = 6;

// ---------------------------------------------------------------------------
// Phase 1: serial recurrence. One thread = one batch row (2048 threads total).
// Embed table in LDS (per-lane dynamic index); transition matrices read via
// wave-uniform global loads so they live in SGPRs (no LDS traffic in the
// serial loop). Class states written to workspace in [t][Bdim][EMB] layout
// => coalesced wave stores each timestep; 48 MB stays resident in the 192 MB
// L2 for phase 2.
// ---------------------------------------------------------------------------
__global__ __launch_bounds__(256) void sticky_scan(
    const int*   __restrict__ x,
    const float* __restrict__ embed,
    const float* __restrict__ hidden_init,
    const float* __restrict__ reset_threshold,
    const float* __restrict__ temperature,
    const float* __restrict__ class_transition,
    const float* __restrict__ reset_transition,
    float*       __restrict__ cls)             // [Tdim][Bdim][EMB]
{
    __shared__ float sE[VOC * EMB + 2];        // +2 pad so v2f read at row 64 is safe
    for (int i = threadIdx.x; i < VOC * EMB + 2; i += blockDim.x)
        sE[i] = (i < VOC * EMB) ? embed[i] : 0.f;
    __syncthreads();

    // Wave-uniform reads -> SGPR-resident constants
    float sC[EMB * EMB], sR[EMB * EMB];
    #pragma unroll
    for (int i = 0; i < EMB * EMB; ++i) { sC[i] = class_transition[i]; sR[i] = reset_transition[i]; }

    const int b = blockIdx.x * blockDim.x + threadIdx.x;   // 0..2047

    float c[EMB];
    #pragma unroll
    for (int j = 0; j < EMB; ++j) c[j] = hidden_init[j];
    float accum = hidden_init[EMB];

    const float thr2 = reset_threshold[0] * reset_threshold[0];
    const float invT = 1.0f / temperature[0];

    const int* xr = x + (size_t)b * Tdim;
    int tok = xr[0];

    for (int t = 0; t < Tdim; ++t) {
        const int tokc = tok;
        if (t + 1 < Tdim) tok = xr[t + 1];                 // prefetch next token

        // Embed gather: row stride 24 B -> 8-byte aligned, three ds_load_b64
        const float* ep = &sE[tokc * EMB];
        const v2f e01 = *(const v2f*)(ep + 0);
        const v2f e23 = *(const v2f*)(ep + 2);
        const v2f e45 = *(const v2f*)(ep + 4);
        float e[EMB] = {e01.x, e01.y, e23.x, e23.y, e45.x, e45.y};

        // Off-chain work: two 6x6 matvecs (SGPR weights fill issue slots)
        float tc[EMB], dl[EMB];
        #pragma unroll
        for (int j = 0; j < EMB; ++j) {
            float tv = 0.f, rv = 0.f;
            #pragma unroll
            for (int k = 0; k < EMB; ++k) {
                tv = fmaf(e[k], sC[k * EMB + j], tv);
                rv = fmaf(e[k], sR[k * EMB + j], rv);
            }
            tc[j] = tv;
            dl[j] = rv - tv;                               // reset_class - transition_class
        }

        // Dependency chain: dot -> relu -> sigmoid -> lerp
        float sim = 0.f;
        #pragma unroll
        for (int k = 0; k < EMB; ++k) sim = fmaf(e[k], c[k], sim);

        const float na    = fmaxf((1.0f + accum) * sim, 0.0f);   // new_accum
        const float z     = (na - thr2) * invT;
        const float reset = 1.0f / (1.0f + __expf(-z));
        accum = na - reset * na;                                 // (1-reset)*new_accum

        #pragma unroll
        for (int j = 0; j < EMB; ++j) c[j] = fmaf(reset, dl[j], tc[j]);

        // 24 B row -> three aligned b64 stores
        float* o = cls + ((size_t)t * Bdim + b) * EMB;
        *(v2f*)(o + 0) = (v2f){c[0], c[1]};
        *(v2f*)(o + 2) = (v2f){c[2], c[3]};
        *(v2f*)(o + 4) = (v2f){c[4], c[5]};
    }
}

// ---------------------------------------------------------------------------
// Phase 2: logits = cls(6) @ output(6x65) via V_WMMA_F32_16X16X4_F32.
// One wave per 16-row tile (rows r = t*Bdim + b; tiles never cross t since
// Bdim % 16 == 0). K=6 -> two WMMAs (K 0..3 and K 4..5 zero padded).
// N=65 -> 5 tiles of 16 cols, last tile valid only at col 64.
//
// The 6x65 output matrix is staged once per block into a zero-padded,
// TRANSPOSED LDS table sOt[80][8] so every B fragment is one unconditional
// aligned ds_load_b64 (no exec-mask branching).
//
// A 16x4 fp32 fragment layout (ISA 7.12.2): lanes 0-15 = M rows, v0 holds
// K = 2*hi, v1 holds K = 2*hi+1 where hi = lane>=16. B mirrors it over N.
// ---------------------------------------------------------------------------
__global__ __launch_bounds__(256) void sticky_logits(
    const float* __restrict__ cls,             // [Tdim][Bdim][EMB]
    const float* __restrict__ Omat,            // [EMB][VOC]
    float*       __restrict__ out)             // [Bdim][Tdim][VOC]
{
    __shared__ float sOt[80 * 8];              // [col][k], zero padded
    for (int i = threadIdx.x; i < 80 * 8; i += blockDim.x) {
        const int col = i >> 3, kr = i & 7;
        sOt[i] = (col < VOC && kr < EMB) ? Omat[kr * VOC + col] : 0.f;
    }
    __syncthreads();

    const int lane = threadIdx.x & 31;
    const int wave = threadIdx.x >> 5;
    const int lo   = lane & 15;
    const int hi   = lane >> 4;                // 0 or 1

    const size_t tile   = (size_t)blockIdx.x * (blockDim.x >> 5) + wave;
    const size_t r_base = tile * 16;
    const int t  = (int)(r_base / Bdim);
    const int b0 = (int)(r_base - (size_t)t * Bdim);

    // A fragments (class-state rows; EMB*4 = 24 B stride -> 8-byte aligned)
    const float* ap = cls + (r_base + (size_t)lo) * EMB;
    v2f a0 = *(const v2f*)(ap + hi * 2);       // K = {0,1} | {2,3}
    v2f a1;
    if (hi) { a1.x = 0.f; a1.y = 0.f; }        // K = {6,7} padding
    else    { a1 = *(const v2f*)(ap + 4); }    // K = {4,5}

    // B fragments: one aligned ds_load_b64 each (k = kr, kr+1 contiguous)
    v2f bf[5][2];
    #pragma unroll
    for (int n = 0; n < 5; ++n) {
        const int col = n * 16 + lo;
        #pragma unroll
        for (int q = 0; q < 2; ++q) {
            const int kr = q * 4 + hi * 2;
            bf[n][q] = *(const v2f*)&sOt[col * 8 + kr];
        }
    }

    v8f acc[5];
    #pragma unroll
    for (int n = 0; n < 5; ++n) {
        v8f z = {0.f, 0.f, 0.f, 0.f, 0.f, 0.f, 0.f, 0.f};
        z = __builtin_amdgcn_wmma_f32_16x16x4_f32(false, a0, false, bf[n][0],
                                                  (short)0, z, false, false);
        z = __builtin_amdgcn_wmma_f32_16x16x4_f32(false, a1, false, bf[n][1],
                                                  (short)0, z, false, false);
        acc[n] = z;
    }

    // D layout: VGPR j, lanes 0-15 -> M=j, lanes 16-31 -> M=j+8, N = lo.
    // Nontemporal: 545 MB result must not evict the 48 MB class states in L2.
    const size_t rowbase = ((size_t)(b0 + hi * 8) * Tdim + t) * VOC;
    #pragma unroll
    for (int n = 0; n < 4; ++n) {              // cols 0..63: always in range
        const int col = n * 16 + lo;
        #pragma unroll
        for (int j = 0; j < 8; ++j)
            __builtin_nontemporal_store(acc[n][j],
                out + rowbase + (size_t)j * (Tdim * VOC) + col);
    }
    if (lo == 0) {                             // tile 4: only col 64 is valid
        #pragma unroll
        for (int j = 0; j < 8; ++j)
            __builtin_nontemporal_store(acc[4][j],
                out + rowbase + (size_t)j * (Tdim * VOC) + 64);
    }
}

// ---------------------------------------------------------------------------
extern "C" void kernel_launch(void* const* d_in, const int* in_sizes, int n_in,
                              void* d_out, int out_size, void* d_ws, size_t ws_size,
                              hipStream_t stream) {
    (void)in_sizes; (void)n_in; (void)out_size; (void)ws_size;

    const int*   x    = (const int*)  d_in[0];
    const float* emb  = (const float*)d_in[1];
    const float* h0   = (const float*)d_in[2];
    const float* thr  = (const float*)d_in[3];
    const float* temp = (const float*)d_in[4];
    const float* ct   = (const float*)d_in[5];
    const float* rt   = (const float*)d_in[6];
    const float* om   = (const float*)d_in[7];
    float*       out  = (float*)d_out;
    float*       cls  = (float*)d_ws;          // needs Bdim*Tdim*EMB*4 = 48 MB

    sticky_scan<<<Bdim / 256, 256, 0, stream>>>(x, emb, h0, thr, temp, ct, rt, cls);

    const int tiles         = (Bdim * Tdim) / 16;   // 131072 row-tiles
    const int wavesPerBlock = 8;                    // 256 threads
    sticky_logits<<<tiles / wavesPerBlock, wavesPerBlock * 32, 0, stream>>>(cls, om, out);
}